// ContrastiveLoss_18502719111626
// MI455X (gfx1250) — compile-verified
//
#include <hip/hip_runtime.h>

// ---------------------------------------------------------------------------
// ContrastiveLoss: out = COF1 * ( ||sum_i m_i * ln_i||^2 - sum_i m_i ) / (2N)
//   m_i  = (row_sum(latent_i) != 0)
//   ln_i = latent_i / max(||latent_i||, EPS)
// N = 8192, d = 64. Pure streaming reduction (2 MB @ 23.3 TB/s ~= 90 ns).
// ---------------------------------------------------------------------------

#define DCOLS   64
#define NBLK    64
#define BLOCK   256          // 8 waves of 32 on gfx1250
#define WAVES_PER_BLOCK (BLOCK / 32)
#define EPS     1e-8f
#define COF1    0.01f

typedef float v2f __attribute__((ext_vector_type(2)));
typedef float v8f __attribute__((ext_vector_type(8)));

// Kernel 1: per-wave row processing, deterministic block partials to ws.
// ws layout: [NBLK * 64] column partials, then [NBLK] mask counts.
__global__ void __launch_bounds__(BLOCK)
cl_reduce_rows(const float* __restrict__ latent, float* __restrict__ part, int nrows) {
    const int lane  = threadIdx.x & 31;
    const int wave  = threadIdx.x >> 5;
    const int gwave = blockIdx.x * WAVES_PER_BLOCK + wave;
    const int nwave = gridDim.x * WAVES_PER_BLOCK;

    float ax = 0.0f, ay = 0.0f, cnt = 0.0f;

    for (int row = gwave; row < nrows; row += nwave) {
        // coalesced: lane l reads columns 2l, 2l+1 (8B/lane, 256B/wave)
        const float2 x = ((const float2*)(latent + (size_t)row * DCOLS))[lane];
        // prefetch next row this wave will touch (global_prefetch_b8)
        if (row + nwave < nrows)
            __builtin_prefetch(latent + (size_t)(row + nwave) * DCOLS + 2 * lane, 0, 0);

        float s = x.x + x.y;                 // row-sum contribution
        float q = fmaf(x.x, x.x, x.y * x.y); // row-sumsq contribution
        #pragma unroll
        for (int off = 16; off > 0; off >>= 1) {   // wave32 butterfly
            s += __shfl_xor(s, off, 32);
            q += __shfl_xor(q, off, 32);
        }
        const float mask  = (s != 0.0f) ? 1.0f : 0.0f;
        const float scale = mask / fmaxf(sqrtf(q), EPS);
        ax = fmaf(scale, x.x, ax);
        ay = fmaf(scale, x.y, ay);
        cnt += mask;                          // identical in every lane
    }

    // Deterministic fixed-order combine of the 8 waves through LDS.
    __shared__ float vsh[WAVES_PER_BLOCK][DCOLS];
    __shared__ float csh[WAVES_PER_BLOCK];
    vsh[wave][2 * lane + 0] = ax;
    vsh[wave][2 * lane + 1] = ay;
    if (lane == 0) csh[wave] = cnt;
    __syncthreads();

    if (threadIdx.x < DCOLS) {
        float t = 0.0f;
        #pragma unroll
        for (int w = 0; w < WAVES_PER_BLOCK; ++w) t += vsh[w][threadIdx.x];
        part[blockIdx.x * DCOLS + threadIdx.x] = t;
    }
    if (threadIdx.x == 0) {
        float t = 0.0f;
        #pragma unroll
        for (int w = 0; w < WAVES_PER_BLOCK; ++w) t += csh[w];
        part[NBLK * DCOLS + blockIdx.x] = t;
    }
}

// Kernel 2: one full wave (EXEC all-ones, required for WMMA).
// Reduces block partials, then ||v||^2 = trace(A x B) via V_WMMA_F32_16X16X4_F32
// with A[m,k] = B[k,n] = v[4m+k] (16x4 / 4x16 reshapes of v).
__global__ void __launch_bounds__(32)
cl_finalize(const float* __restrict__ part, float* __restrict__ out, int nrows) {
    const int lane = threadIdx.x;   // exactly one wave32

    float vx = 0.0f, vy = 0.0f, cnt = 0.0f;
    for (int b = 0; b < NBLK; ++b) {
        vx  += part[b * DCOLS + 2 * lane + 0];
        vy  += part[b * DCOLS + 2 * lane + 1];
        cnt += part[NBLK * DCOLS + b];
    }
    // lane l owns v[2l], v[2l+1].
    // A-operand layout (16x4 f32): lanes 0-15 hold {v[4m],v[4m+1]} (m=lane),
    // lanes 16-31 hold {v[4m+2],v[4m+3]} (m=lane-16). Source lane:
    const int src = (lane < 16) ? (2 * lane) : (2 * (lane - 16) + 1);
    v2f a;
    a.x = __shfl(vx, src, 32);
    a.y = __shfl(vy, src, 32);

    v8f c = {};
    // D[m,n] = sum_k v[4m+k]*v[4n+k]  =>  trace(D) = ||v||^2
    c = __builtin_amdgcn_wmma_f32_16x16x4_f32(
            /*neg_a=*/false, a, /*neg_b=*/false, a,
            /*c_mod=*/(short)0, c, /*reuse_a=*/false, /*reuse_b=*/false);

    // Diagonal of 16x16 f32 C/D: D[g,g] @ (vgpr g, lane g) for g<8,
    // D[8+g,8+g] @ (vgpr g, lane 24+g).
    float t = 0.0f;
    #pragma unroll
    for (int g = 0; g < 8; ++g) {
        const float dg = c[g];
        if (lane == g || lane == 24 + g) t += dg;
    }
    #pragma unroll
    for (int off = 16; off > 0; off >>= 1) t += __shfl_xor(t, off, 32);

    if (lane == 0) {
        const float total = t - cnt;          // minus self-similarity correction
        out[0] = COF1 * total / (2.0f * (float)nrows);
    }
}

extern "C" void kernel_launch(void* const* d_in, const int* in_sizes, int n_in,
                              void* d_out, int out_size, void* d_ws, size_t ws_size,
                              hipStream_t stream) {
    (void)n_in; (void)out_size; (void)ws_size;
    const float* latent = (const float*)d_in[0];
    const int nrows = in_sizes[0] / DCOLS;    // 8192
    float* part = (float*)d_ws;               // (NBLK*64 + NBLK) floats = 16.6 KB

    cl_reduce_rows<<<NBLK, BLOCK, 0, stream>>>(latent, part, nrows);
    cl_finalize<<<1, 32, 0, stream>>>(part, (float*)d_out, nrows);
}